// CBPoolMax2d_65111704207913
// MI455X (gfx1250) — compile-verified
//
#include <hip/hip_runtime.h>

// Problem constants (match reference: C=256, H=512, W=512, floor-mode pool -> 256x256)
constexpr int Cc  = 256;
constexpr int Hh  = 512;
constexpr int Ww  = 512;
constexpr int OH  = Hh / 2;
constexpr int OW  = Ww / 2;

// Native clang vector type: accepted by __builtin_nontemporal_* and lowers to
// 128-bit global memory ops.
typedef float v4f __attribute__((ext_vector_type(4)));

// ---------------------------------------------------------------------------
// Kernel 1: copy persistent state -> output, 128-bit vectorized, non-temporal
// loads/stores (state is streamed exactly once; keep L2 free for the input
// tensor which the scatter pass re-reads with heavy channel-wise reuse).
// ---------------------------------------------------------------------------
__global__ void copy_state_b128(const v4f* __restrict__ src,
                                v4f* __restrict__ dst, int n4) {
    int i = blockIdx.x * blockDim.x + threadIdx.x;
    if (i < n4) {
        v4f v = __builtin_nontemporal_load(&src[i]);      // global_load_b128 th:NT
        __builtin_nontemporal_store(v, &dst[i]);          // global_store_b128 th:NT
    }
}

// ---------------------------------------------------------------------------
// Kernel 2: sparse 2x2 max-pool recompute + scatter.
// grid.x = K (one block per changed pixel), blockDim.x = 256 (= channels,
// i.e. 8 wave32s per block on CDNA5).
// The per-block index load is uniform -> SMEM s_load_b32.
// Each thread: two aligned float2 loads (2x2 window rows), 3 v_max, 1 store.
// ---------------------------------------------------------------------------
__global__ void scatter_pool2x2(const float* __restrict__ in,
                                const int* __restrict__ change,
                                float* __restrict__ out, int K) {
    const int k = blockIdx.x;
    const int c = threadIdx.x;

    // gfx1250 prefetch path: warm the next block's index line (speculative).
    if (k + 1 < K && c == 0) {
        __builtin_prefetch(&change[k + 1], 0, 0);         // global_prefetch_b8
    }

    const int idx = change[k];            // uniform across block -> scalar load
    const int y   = idx >> 9;             // idx / W   (W = 512)
    const int x   = idx & (Ww - 1);       // idx % W
    const int iy  = y & ~1;               // top-left of 2x2 window
    const int ix  = x & ~1;               // even -> float2 is 8B aligned

    const float* base = in + (size_t)c * (Hh * Ww) + (size_t)iy * Ww + ix;
    const float2 r0 = *(const float2*)(base);             // global_load_b64
    const float2 r1 = *(const float2*)(base + Ww);        // global_load_b64

    const float v = fmaxf(fmaxf(r0.x, r0.y), fmaxf(r1.x, r1.y));

    out[(size_t)c * (OH * OW) + (size_t)(iy >> 1) * OW + (ix >> 1)] = v;
}

// ---------------------------------------------------------------------------
// Launch. Inputs (setup_inputs order):
//   d_in[0] = input        float32 [1,256,512,512]
//   d_in[1] = outputState  float32 [1,256,256,256]
//   d_in[2] = changeIndexes int32  [K]
// d_out = float32 [1,256,256,256]
// Copy must complete before scatter (scatter overwrites copied pixels):
// stream ordering between the two kernels provides that.
// ---------------------------------------------------------------------------
extern "C" void kernel_launch(void* const* d_in, const int* in_sizes, int n_in,
                              void* d_out, int out_size, void* d_ws, size_t ws_size,
                              hipStream_t stream) {
    const float* inp    = (const float*)d_in[0];
    const float* state  = (const float*)d_in[1];
    const int*   change = (const int*)d_in[2];
    float*       out    = (float*)d_out;

    const int n4 = out_size / 4;                           // 16M floats -> 4M float4
    copy_state_b128<<<(n4 + 255) / 256, 256, 0, stream>>>(
        (const v4f*)state, (v4f*)out, n4);

    const int K = in_sizes[2];
    scatter_pool2x2<<<K, 256, 0, stream>>>(inp, change, out, K);
}